// SimpleGNN_52767968198664
// MI455X (gfx1250) — compile-verified
//
#include <hip/hip_runtime.h>
#include <hip/hip_bf16.h>

// ---------------------------------------------------------------------------
// 2-layer GAT (PyG GATConv semantics) for MI455X / gfx1250.
//  - Dense projections run on the WMMA pipe with V_WMMA_F32_16X16X4_F32
//    (exact fp32 math, RNE accumulate -> matches the f32 reference).
//  - Edge softmax/aggregation is atomic-based (segment max / sum / weighted
//    scatter-add), which is the HBM/atomic-bound part of the workload.
// ---------------------------------------------------------------------------

typedef float v2f __attribute__((ext_vector_type(2)));
typedef float v8f __attribute__((ext_vector_type(8)));

#define NEG_FLT_MAX (-3.402823466e38f)

// --------------------------- utility kernels -------------------------------

__global__ void fill_kernel(float* __restrict__ p, float v, int n) {
    int i = blockIdx.x * blockDim.x + threadIdx.x;
    if (i < n) p[i] = v;
}

// float atomic max via int punning (valid for mixed signs; init = -FLT_MAX)
__device__ __forceinline__ void atomicMaxFloat(float* addr, float val) {
    if (val >= 0.0f) {
        atomicMax((int*)addr, __float_as_int(val));
    } else {
        atomicMin((unsigned int*)addr, (unsigned int)__float_as_int(val));
    }
}

// --------------------------- WMMA fp32 GEMM --------------------------------
// C[M x 64] = A[M x K] * B[K x 64], all fp32 row-major. One wave computes a
// 16-row x 64-col strip using 4 v8f accumulators; K consumed 4 at a time by
// V_WMMA_F32_16X16X4_F32.
//
// Per the CDNA5 VGPR layouts (ISA 7.12.2):
//  A 16x4:  lane l (l<16): M=l, holds K = 0,1 ; lane l+16: M=l, K = 2,3
//  B 4x16:  VGPR0 = row K=(2*half), VGPR1 = row K=(2*half+1), N = lane&15
//  C 16x16: acc[v] -> row = v + 8*half, col = lane&15
__global__ void __launch_bounds__(256)
wmma_gemm_f32(const float* __restrict__ A, const float* __restrict__ B,
              float* __restrict__ C, int M, int K) {
    const int wave = blockIdx.x * 8 + (threadIdx.x >> 5);
    const int lane = threadIdx.x & 31;
    const int half = lane >> 4;   // selects K pair
    const int lid  = lane & 15;
    const int row0 = wave * 16;
    if (row0 >= M) return;        // uniform per wave: EXEC stays all-ones

    v8f acc0 = {}; v8f acc1 = {}; v8f acc2 = {}; v8f acc3 = {};

    int arow = row0 + lid;
    if (arow >= M) arow = M - 1;  // redundant lane, result masked on store
    const float* __restrict__ ap = A + (size_t)arow * K + half * 2;

    for (int kk = 0; kk < K; kk += 4) {
        v2f a;
        a.x = ap[kk];
        a.y = ap[kk + 1];
        const float* __restrict__ bp = B + (size_t)(kk + half * 2) * 64 + lid;
        v2f b0; b0.x = bp[0];  b0.y = bp[64];
        v2f b1; b1.x = bp[16]; b1.y = bp[80];
        v2f b2; b2.x = bp[32]; b2.y = bp[96];
        v2f b3; b3.x = bp[48]; b3.y = bp[112];
        acc0 = __builtin_amdgcn_wmma_f32_16x16x4_f32(false, a, false, b0, (short)0, acc0, false, false);
        acc1 = __builtin_amdgcn_wmma_f32_16x16x4_f32(false, a, false, b1, (short)0, acc1, false, false);
        acc2 = __builtin_amdgcn_wmma_f32_16x16x4_f32(false, a, false, b2, (short)0, acc2, false, false);
        acc3 = __builtin_amdgcn_wmma_f32_16x16x4_f32(false, a, false, b3, (short)0, acc3, false, false);
    }

#pragma unroll
    for (int v = 0; v < 8; ++v) {
        int row = row0 + v + half * 8;
        if (row < M) {
            float* __restrict__ cp = C + (size_t)row * 64 + lid;
            cp[0]  = acc0[v];
            cp[16] = acc1[v];
            cp[32] = acc2[v];
            cp[48] = acc3[v];
        }
    }
}

// ------------------------ attention coefficients ---------------------------

// layer 1: a_src/a_dst[n,h] = sum_c h1[n,h*8+c] * att[h*8+c]   (H=8, C=8)
__global__ void att1_kernel(const float* __restrict__ h1,
                            const float* __restrict__ att_src,
                            const float* __restrict__ att_dst,
                            float* __restrict__ asrc, float* __restrict__ adst,
                            int N) {
    int i = blockIdx.x * blockDim.x + threadIdx.x;   // i = n*8 + h
    if (i >= N * 8) return;
    int h = i & 7;
    int n = i >> 3;
    const float* hp = h1 + (size_t)n * 64 + h * 8;
    const float* as = att_src + h * 8;
    const float* ad = att_dst + h * 8;
    float ss = 0.f, dd = 0.f;
#pragma unroll
    for (int c = 0; c < 8; ++c) { ss += hp[c] * as[c]; dd += hp[c] * ad[c]; }
    asrc[i] = ss;
    adst[i] = dd;
}

// layer 2: single head over 64 channels
__global__ void att2_kernel(const float* __restrict__ h2,
                            const float* __restrict__ att_src,
                            const float* __restrict__ att_dst,
                            float* __restrict__ asrc, float* __restrict__ adst,
                            int N) {
    int n = blockIdx.x * blockDim.x + threadIdx.x;
    if (n >= N) return;
    const float* hp = h2 + (size_t)n * 64;
    float ss = 0.f, dd = 0.f;
#pragma unroll
    for (int c = 0; c < 64; ++c) { ss += hp[c] * att_src[c]; dd += hp[c] * att_dst[c]; }
    asrc[n] = ss;
    adst[n] = dd;
}

// --------------------------- edge-parallel passes --------------------------

__device__ __forceinline__ void edge_ids(const int* __restrict__ src,
                                         const int* __restrict__ dst,
                                         int e, int E0, int& s, int& d) {
    if (e < E0) { s = src[e]; d = dst[e]; }
    else        { s = e - E0; d = e - E0; }   // self loops appended
}

template <int H>
__global__ void edge_max_kernel(const int* __restrict__ src, const int* __restrict__ dst,
                                int E0, int Etot,
                                const float* __restrict__ asrc,
                                const float* __restrict__ adst,
                                float* __restrict__ m) {
    int e = blockIdx.x * blockDim.x + threadIdx.x;
    if (e >= Etot) return;
    int s, d; edge_ids(src, dst, e, E0, s, d);
#pragma unroll
    for (int h = 0; h < H; ++h) {
        float v = asrc[s * H + h] + adst[d * H + h];
        v = v > 0.f ? v : 0.2f * v;                 // leaky_relu(0.2)
        atomicMaxFloat(&m[d * H + h], v);
    }
}

template <int H>
__global__ void edge_sum_kernel(const int* __restrict__ src, const int* __restrict__ dst,
                                int E0, int Etot,
                                const float* __restrict__ asrc,
                                const float* __restrict__ adst,
                                const float* __restrict__ m,
                                float* __restrict__ ssum) {
    int e = blockIdx.x * blockDim.x + threadIdx.x;
    if (e >= Etot) return;
    int s, d; edge_ids(src, dst, e, E0, s, d);
#pragma unroll
    for (int h = 0; h < H; ++h) {
        float v = asrc[s * H + h] + adst[d * H + h];
        v = v > 0.f ? v : 0.2f * v;
        atomicAdd(&ssum[d * H + h], expf(v - m[d * H + h]));
    }
}

template <int H, int C>
__global__ void edge_agg_kernel(const int* __restrict__ src, const int* __restrict__ dst,
                                int E0, int Etot,
                                const float* __restrict__ asrc,
                                const float* __restrict__ adst,
                                const float* __restrict__ m,
                                const float* __restrict__ ssum,
                                const float* __restrict__ hfeat,
                                float* __restrict__ out) {
    int e = blockIdx.x * blockDim.x + threadIdx.x;
    if (e >= Etot) return;
    int s, d; edge_ids(src, dst, e, E0, s, d);
    float alpha[H];
#pragma unroll
    for (int h = 0; h < H; ++h) {
        float v = asrc[s * H + h] + adst[d * H + h];
        v = v > 0.f ? v : 0.2f * v;
        alpha[h] = expf(v - m[d * H + h]) / (ssum[d * H + h] + 1e-16f);
    }
    const float* __restrict__ hp = hfeat + (size_t)s * (H * C);
    float* __restrict__ op = out + (size_t)d * (H * C);
#pragma unroll
    for (int h = 0; h < H; ++h) {
        float al = alpha[h];
#pragma unroll
        for (int c = 0; c < C; ++c)
            atomicAdd(&op[h * C + c], hp[h * C + c] * al);
    }
}

// ------------------------------ epilogues ----------------------------------

// h_elu = elu(out1 + b1)   (64 channels)
__global__ void elu_bias_kernel(const float* __restrict__ in,
                                const float* __restrict__ bias,
                                float* __restrict__ out, int total) {
    int i = blockIdx.x * blockDim.x + threadIdx.x;
    if (i >= total) return;
    float v = in[i] + bias[i & 63];
    out[i] = v > 0.f ? v : (expf(v) - 1.0f);
}

// y[n] = sum_c (out2[n,c] + b2[c]) * linW[c] + linb
__global__ void final_kernel(const float* __restrict__ out2,
                             const float* __restrict__ b2,
                             const float* __restrict__ linW,
                             const float* __restrict__ linb,
                             float* __restrict__ y, int N) {
    int n = blockIdx.x * blockDim.x + threadIdx.x;
    if (n >= N) return;
    const float* op = out2 + (size_t)n * 64;
    float acc = 0.f;
#pragma unroll
    for (int c = 0; c < 64; ++c) acc += (op[c] + b2[c]) * linW[c];
    y[n] = acc + linb[0];
}

// ------------------------------- launcher ----------------------------------

extern "C" void kernel_launch(void* const* d_in, const int* in_sizes, int n_in,
                              void* d_out, int out_size, void* d_ws, size_t ws_size,
                              hipStream_t stream) {
    const float* x        = (const float*)d_in[0];
    const int*   ei       = (const int*)d_in[1];   // int32 on device (jax default x64 off)
    const float* W1       = (const float*)d_in[2];
    const float* att_src1 = (const float*)d_in[3];
    const float* att_dst1 = (const float*)d_in[4];
    const float* b1       = (const float*)d_in[5];
    const float* W2       = (const float*)d_in[6];
    const float* att_src2 = (const float*)d_in[7];
    const float* att_dst2 = (const float*)d_in[8];
    const float* b2       = (const float*)d_in[9];
    const float* linW     = (const float*)d_in[10];
    const float* linb     = (const float*)d_in[11];
    float*       y        = (float*)d_out;

    const int N    = in_sizes[0] / 128;   // 50000
    const int E0   = in_sizes[1] / 2;     // 1600000
    const int Etot = E0 + N;              // + self loops
    const int* src = ei;
    const int* dst = ei + E0;

    // workspace layout (fp32)
    float* ws    = (float*)d_ws;
    float* h1    = ws; ws += (size_t)N * 64;   // reused as h_elu after layer-1 agg
    float* asrc1 = ws; ws += (size_t)N * 8;
    float* adst1 = ws; ws += (size_t)N * 8;
    float* m1    = ws; ws += (size_t)N * 8;
    float* s1    = ws; ws += (size_t)N * 8;
    float* out1  = ws; ws += (size_t)N * 64;
    float* h2    = ws; ws += (size_t)N * 64;
    float* asrc2 = ws; ws += (size_t)N;
    float* adst2 = ws; ws += (size_t)N;
    float* m2    = ws; ws += (size_t)N;
    float* s2    = ws; ws += (size_t)N;
    float* out2  = ws; ws += (size_t)N * 64;

    const int B = 256;
    auto g = [](int n, int b) { return (n + b - 1) / b; };
    const int gemm_blocks = g(g(N, 16), 8);   // 8 waves/block, 16 rows/wave
    const int edge_blocks = g(Etot, B);

    // ---- init accumulators (every call: deterministic under graph replay)
    fill_kernel<<<g(N * 8, B), B, 0, stream>>>(m1, NEG_FLT_MAX, N * 8);
    fill_kernel<<<g(N * 8, B), B, 0, stream>>>(s1, 0.f, N * 8);
    fill_kernel<<<g(N * 64, B), B, 0, stream>>>(out1, 0.f, N * 64);
    fill_kernel<<<g(N, B), B, 0, stream>>>(m2, NEG_FLT_MAX, N);
    fill_kernel<<<g(N, B), B, 0, stream>>>(s2, 0.f, N);
    fill_kernel<<<g(N * 64, B), B, 0, stream>>>(out2, 0.f, N * 64);

    // ---- layer 1: projection (WMMA), attention, segment softmax, aggregate
    wmma_gemm_f32<<<gemm_blocks, B, 0, stream>>>(x, W1, h1, N, 128);
    att1_kernel<<<g(N * 8, B), B, 0, stream>>>(h1, att_src1, att_dst1, asrc1, adst1, N);
    edge_max_kernel<8><<<edge_blocks, B, 0, stream>>>(src, dst, E0, Etot, asrc1, adst1, m1);
    edge_sum_kernel<8><<<edge_blocks, B, 0, stream>>>(src, dst, E0, Etot, asrc1, adst1, m1, s1);
    edge_agg_kernel<8, 8><<<edge_blocks, B, 0, stream>>>(src, dst, E0, Etot, asrc1, adst1, m1, s1, h1, out1);
    elu_bias_kernel<<<g(N * 64, B), B, 0, stream>>>(out1, b1, h1, N * 64);  // h1 <- elu(out1+b1)

    // ---- layer 2: projection (WMMA), attention, segment softmax, aggregate
    wmma_gemm_f32<<<gemm_blocks, B, 0, stream>>>(h1, W2, h2, N, 64);
    att2_kernel<<<g(N, B), B, 0, stream>>>(h2, att_src2, att_dst2, asrc2, adst2, N);
    edge_max_kernel<1><<<edge_blocks, B, 0, stream>>>(src, dst, E0, Etot, asrc2, adst2, m2);
    edge_sum_kernel<1><<<edge_blocks, B, 0, stream>>>(src, dst, E0, Etot, asrc2, adst2, m2, s2);
    edge_agg_kernel<1, 64><<<edge_blocks, B, 0, stream>>>(src, dst, E0, Etot, asrc2, adst2, m2, s2, h2, out2);

    // ---- readout
    final_kernel<<<g(N, B), B, 0, stream>>>(out2, b2, linW, linb, y, N);
}